// OuterProduct_16810501997099
// MI455X (gfx1250) — compile-verified
//
#include <hip/hip_runtime.h>
#include <cstdint>

typedef __attribute__((ext_vector_type(2))) float v2f;
typedef __attribute__((ext_vector_type(4))) float v4f;
typedef __attribute__((ext_vector_type(8))) float v8f;

constexpr int B_ROWS         = 8192;
constexpr int L_COLS         = 8192;
constexpr int ROWS_PER_BLOCK = 16;           // one WMMA M-tile
constexpr int TILE_K         = 512;          // columns staged to LDS per stage
constexpr int PITCH          = TILE_K + 4;   // 516: conflict-free WMMA A reads
constexpr int NSTAGES        = L_COLS / TILE_K;   // 16
constexpr int THREADS1       = 256;          // 8 waves (wave32)
constexpr int WAVES1         = THREADS1 / 32;
constexpr int KSLICE         = TILE_K / WAVES1;   // 64 cols per wave -> 16 WMMAs

// Issue 8 async global->LDS b128 loads covering a 16 x 512 f32 tile.
// Per wave this is exactly 8 ASYNCcnt increments.
__device__ __forceinline__ void stage_load(const float* __restrict__ gbase,
                                           int st, float* tbase, int t) {
#pragma unroll
  for (int i = 0; i < 8; ++i) {
    const int f   = t + i * THREADS1;       // 2048 float4s total
    const int row = f >> 7;                 // 128 float4 per row
    const int col = (f & 127) << 2;         // float index within row
    const unsigned voff = (unsigned)((row * L_COLS + st * TILE_K + col) * 4u);
    const unsigned lds  = (unsigned)(uintptr_t)(tbase + row * PITCH + col);
    asm volatile("global_load_async_to_lds_b128 %0, %1, %2"
                 :: "v"(lds), "v"(voff), "s"(gbase)
                 : "memory");
  }
}

// s_out[n] = sum_j a[n, j]   (16 rows per block, WMMA row-sum via B = ones)
__global__ __launch_bounds__(THREADS1)
void rowsum_wmma_kernel(const float* __restrict__ a, float* __restrict__ s_out) {
  __shared__ __align__(16) float tile[2][ROWS_PER_BLOCK][PITCH];
  __shared__ float partial[WAVES1 * ROWS_PER_BLOCK];

  const int row0 = blockIdx.x * ROWS_PER_BLOCK;
  const int t    = threadIdx.x;
  const int wave = t >> 5;
  const int lane = t & 31;

  const float* gbase = a + (size_t)row0 * L_COLS;

  // WMMA A-fragment addressing (16x4 f32): lanes 0-15 -> K={k,k+1},
  // lanes 16-31 -> K={k+2,k+3}, M = lane & 15.
  const int arow = lane & 15;
  const int koff = (lane >> 4) << 1;

  const v2f ones = {1.0f, 1.0f};
  v8f c = {0.f, 0.f, 0.f, 0.f, 0.f, 0.f, 0.f, 0.f};

  stage_load(gbase, 0, &tile[0][0][0], t);

  int buf = 0;
  for (int st = 0; st < NSTAGES; ++st) {
    if (st + 1 < NSTAGES) {
      stage_load(gbase, st + 1, &tile[buf ^ 1][0][0], t);
      asm volatile("s_wait_asynccnt 0x8" ::: "memory");   // current stage resident
    } else {
      asm volatile("s_wait_asynccnt 0x0" ::: "memory");
    }
    __syncthreads();

    const float* trow = &tile[buf][arow][0];
    const int kbase = wave * KSLICE;
#pragma unroll
    for (int j = 0; j < KSLICE / 4; ++j) {
      const v2f aa = *(const v2f*)(trow + kbase + 4 * j + koff);
      // D[m,n] += sum_k A[m,k] * 1  -> row sums replicated across N
      c = __builtin_amdgcn_wmma_f32_16x16x4_f32(
              false, aa, false, ones, (short)0, c, false, false);
    }
    __syncthreads();   // safe to overwrite tile[buf] next iteration
    buf ^= 1;
  }

  // D layout: VGPR r, lanes 0-15 -> M=r ; lanes 16-31 -> M=r+8 (N = lane&15).
  if (lane == 0) {
#pragma unroll
    for (int r = 0; r < 8; ++r) partial[wave * 16 + r] = c[r];
  }
  if (lane == 16) {
#pragma unroll
    for (int r = 0; r < 8; ++r) partial[wave * 16 + 8 + r] = c[r];
  }
  __syncthreads();

  if (t < ROWS_PER_BLOCK) {
    float sum = 0.f;
#pragma unroll
    for (int w = 0; w < WAVES1; ++w) sum += partial[w * 16 + t];
    s_out[row0 + t] = sum;
  }
}

// out[n, i] = b[n, i] * s[n]   — pure NT stream, 16384 blocks
constexpr int THREADS2       = 256;
constexpr int F4_PER_THREAD  = 4;
constexpr int COLS_PER_BLOCK = THREADS2 * F4_PER_THREAD * 4;  // 4096

__global__ __launch_bounds__(THREADS2)
void scale_rows_kernel(const float* __restrict__ b,
                       const float* __restrict__ s_in,
                       float* __restrict__ out) {
  const int row = blockIdx.y;
  const float s = s_in[row];
  const size_t base = (size_t)row * L_COLS + (size_t)blockIdx.x * COLS_PER_BLOCK;
  const v4f* b4 = (const v4f*)(b + base);
  v4f*       o4 = (v4f*)(out + base);
#pragma unroll
  for (int i = 0; i < F4_PER_THREAD; ++i) {
    const int idx = threadIdx.x + i * THREADS2;
    v4f v = __builtin_nontemporal_load(b4 + idx);
    v *= s;
    __builtin_nontemporal_store(v, o4 + idx);
  }
}

extern "C" void kernel_launch(void* const* d_in, const int* in_sizes, int n_in,
                              void* d_out, int out_size, void* d_ws, size_t ws_size,
                              hipStream_t stream) {
  const float* a = (const float*)d_in[0];
  const float* b = (const float*)d_in[1];
  float* out = (float*)d_out;
  float* s   = (float*)d_ws;   // 8192 floats = 32 KiB, fully overwritten each call

  rowsum_wmma_kernel<<<B_ROWS / ROWS_PER_BLOCK, THREADS1, 0, stream>>>(a, s);

  dim3 g2(L_COLS / COLS_PER_BLOCK, B_ROWS);
  scale_rows_kernel<<<g2, THREADS2, 0, stream>>>(b, s, out);
}